// Patchify_66005057405681
// MI455X (gfx1250) — compile-verified
//
#include <hip/hip_runtime.h>

typedef __attribute__((ext_vector_type(2))) float v2f;
typedef __attribute__((ext_vector_type(4))) float v4f;
typedef __attribute__((ext_vector_type(8))) float v8f;
typedef int v4i_t __attribute__((vector_size(16)));   // matches builtin param type

#define KDIM       768   // 3*16*16
#define NDIM       768
#define M_BLK      128
#define N_BLK      128
#define K_CHUNK    32
#define LDS_STRIDE 36    // 32 + 4 pad: 16B-aligned b128, conflict-free fragment reads
#define C_STRIDE   132   // 128 + 4 pad for epilogue tile
#define NCHUNK     (KDIM / K_CHUNK)                        // 24
#define BUF_FLOATS (M_BLK * LDS_STRIDE + N_BLK * LDS_STRIDE)  // 9216 floats / buffer

#if __has_builtin(__builtin_amdgcn_global_load_async_to_lds_b128) && \
    __has_builtin(__builtin_amdgcn_s_wait_asynccnt)
#define USE_ASYNC 1
#else
#define USE_ASYNC 0
#endif

// Hilbert d -> (x, y) on a 16x16 grid (classic d2xy), matches reference.
__device__ __forceinline__ void hilbert_d2xy(int d, int& hx, int& hy) {
  int x = 0, y = 0, t = d;
  #pragma unroll
  for (int s = 1; s < 16; s <<= 1) {
    int rx = 1 & (t >> 1);
    int ry = 1 & (t ^ rx);
    if (ry == 0) {
      if (rx == 1) { x = s - 1 - x; y = s - 1 - y; }
      int tmp = x; x = y; y = tmp;
    }
    x += s * rx;
    y += s * ry;
    t >>= 2;
  }
  hx = x; hy = y;
}

// 256 threads = 8 waves. Workgroup computes a 128(M) x 128(N) output block.
// Wave w owns M-tile row w (16 patches) x all 8 N-tiles (8 v8f accumulators).
// K streamed in 32-wide chunks through double-buffered LDS via async DMA.
__global__ __launch_bounds__(256) void patchify_wmma_f32(
    const float* __restrict__ x,      // [128,3,256,256]
    const float* __restrict__ Wmat,   // [768,3,16,16] == [768][768k] row-major
    const float* __restrict__ bias,   // [768]
    float* __restrict__ out)          // [128,256,768]
{
  __shared__ float smem[2 * BUF_FLOATS];   // 73,728 B; reused as 128x132 C tile

  const int tid  = threadIdx.x;
  const int lane = tid & 31;
  const int wave = tid >> 5;
  const int mblk = blockIdx.x;        // 0..255 : 128-patch group (2 per image)
  const int nblk = blockIdx.y;        // 0..5
  const int img  = mblk >> 1;
  const int p0   = (mblk & 1) << 7;
  const int n0   = nblk * N_BLK;

  // ---- per-thread load mapping: 1024 A-quads + 1024 B-quads per chunk, 4+4 per thread
  const int mbase = tid >> 3;         // patch / W-row index for this thread's quads
  const int sub   = tid & 7;
  const int r_sel = sub >> 2;         // which of the 2 pixel-rows in this chunk
  const int q4    = (sub & 3) << 2;   // pixel quad offset within row
  const int kq    = sub << 2;         // B: k offset within chunk (0..28)

  int prow[4], pcol[4];
  #pragma unroll
  for (int j = 0; j < 4; ++j) {
    int hx, hy;
    hilbert_d2xy(p0 + mbase + 32 * j, hx, hy);
    prow[j] = hx * 16;
    pcol[j] = hy * 16;
  }
  const float* xb = x + (size_t)img * 3 * 256 * 256;

  v8f acc[8];
  #pragma unroll
  for (int j = 0; j < 8; ++j) { v8f z = {}; acc[j] = z; }

  const int m0    = wave * 16;
  const int mrow  = m0 + (lane & 15);
  const int khalf = (lane >> 4) << 1;   // 0 or 2: which K pair this half-wave holds

  // Stage k-chunk kc into buffer (A then B), 8 x b128 per thread.
  auto stage = [&](int kc, float* Ab, float* Bb) {
    const int fr = kc * 2 + r_sel;      // flattened (c,py) row, 0..47
    const int c  = fr >> 4;
    const int py = fr & 15;
#if USE_ASYNC
    #pragma unroll
    for (int j = 0; j < 4; ++j) {
      const float* ga = xb + ((size_t)c * 256 + prow[j] + py) * 256 + pcol[j] + q4;
      float*       la = Ab + (mbase + 32 * j) * LDS_STRIDE + r_sel * 16 + q4;
      __builtin_amdgcn_global_load_async_to_lds_b128((v4i_t*)ga, (v4i_t*)la, 0, 0);
      const float* gb = Wmat + (size_t)(n0 + mbase + 32 * j) * KDIM + kc * K_CHUNK + kq;
      float*       lb = Bb + (mbase + 32 * j) * LDS_STRIDE + kq;
      __builtin_amdgcn_global_load_async_to_lds_b128((v4i_t*)gb, (v4i_t*)lb, 0, 0);
    }
#else
    v4f aval[4], bval[4];
    #pragma unroll
    for (int j = 0; j < 4; ++j) {
      aval[j] = *(const v4f*)(xb + ((size_t)c * 256 + prow[j] + py) * 256 + pcol[j] + q4);
      bval[j] = *(const v4f*)(Wmat + (size_t)(n0 + mbase + 32 * j) * KDIM + kc * K_CHUNK + kq);
    }
    #pragma unroll
    for (int j = 0; j < 4; ++j) {
      *(v4f*)&Ab[(mbase + 32 * j) * LDS_STRIDE + r_sel * 16 + q4] = aval[j];
      *(v4f*)&Bb[(mbase + 32 * j) * LDS_STRIDE + kq]              = bval[j];
    }
#endif
  };

  stage(0, smem, smem + M_BLK * LDS_STRIDE);
#if USE_ASYNC
  __builtin_amdgcn_s_wait_asynccnt(0);
#endif
  __syncthreads();

  for (int kc = 0; kc < NCHUNK; ++kc) {
    const int cur = kc & 1;
    if (kc + 1 < NCHUNK) {
      float* An = smem + (cur ^ 1) * BUF_FLOATS;   // freed by barrier at end of kc-1
      stage(kc + 1, An, An + M_BLK * LDS_STRIDE);
    }

    const float* Ab = smem + cur * BUF_FLOATS;
    const float* Bb = Ab + M_BLK * LDS_STRIDE;
    #pragma unroll
    for (int ks = 0; ks < K_CHUNK; ks += 4) {
      v2f afrag = *(const v2f*)&Ab[mrow * LDS_STRIDE + ks + khalf];
      #pragma unroll
      for (int j = 0; j < 8; ++j) {
        v2f bfrag = *(const v2f*)&Bb[(j * 16 + (lane & 15)) * LDS_STRIDE + ks + khalf];
        acc[j] = __builtin_amdgcn_wmma_f32_16x16x4_f32(
            false, afrag, false, bfrag, (short)0, acc[j], false, false);
      }
    }

#if USE_ASYNC
    __builtin_amdgcn_s_wait_asynccnt(0);   // next chunk fully landed in LDS
#endif
    __syncthreads();                        // all waves done with cur buffer
  }

  // ---- epilogue: bias add, transpose through LDS, coalesced NT b128 stores ----
  // C layout per wave: VGPR v -> row m0 + v + 8*(lane/16), col = j*16 + lane%16.
  float* C = smem;                          // 128 x C_STRIDE (fits in smem)
  #pragma unroll
  for (int j = 0; j < 8; ++j) {
    const int nl = j * 16 + (lane & 15);
    const float bv = bias[n0 + nl];
    #pragma unroll
    for (int v = 0; v < 8; ++v) {
      C[(m0 + ((lane >> 4) << 3) + v) * C_STRIDE + nl] = acc[j][v] + bv;
    }
  }
  __syncthreads();

  const size_t out_base = (size_t)mblk * M_BLK * NDIM + n0;
  #pragma unroll
  for (int i = 0; i < 16; ++i) {
    const int qi = tid + 256 * i;
    const int r  = qi >> 5;                 // row 0..127
    const int c4 = (qi & 31) << 2;          // col quad 0..124
    v4f val = *(const v4f*)&C[r * C_STRIDE + c4];
    __builtin_nontemporal_store(val, (v4f*)(out + out_base + (size_t)r * NDIM + c4));
  }
}

extern "C" void kernel_launch(void* const* d_in, const int* in_sizes, int n_in,
                              void* d_out, int out_size, void* d_ws, size_t ws_size,
                              hipStream_t stream) {
  const float* x    = (const float*)d_in[0];   // [128,3,256,256]
  const float* Wmat = (const float*)d_in[1];   // [768,3,16,16]
  const float* bias = (const float*)d_in[2];   // [768]
  float* out = (float*)d_out;                  // [128,256,768]

  dim3 grid(32768 / M_BLK, NDIM / N_BLK);      // (256, 6)
  dim3 block(256);
  patchify_wmma_f32<<<grid, block, 0, stream>>>(x, Wmat, bias, out);
}